// ContrastiveKernelLoss_21139829030980
// MI455X (gfx1250) — compile-verified
//
#include <hip/hip_runtime.h>
#include <math.h>

// ContrastiveKernelLoss for (128,128,3,3) f32 kernels on gfx1250.
//
// Math restructuring: reflect-padded 3x3 gaussian blur of a 3x3 image is a
// constant 9x9 linear map W (separable W1 (x) W1).  Per-filter terms
// mu = W x and bxx = W (x*x) are precomputed; the only pairwise term,
// blur(x_i * x_j)[p] = sum_k W[p][k] x_i[k] x_j[k], is a per-pixel Gram
// matrix (diag(w_p) X) X^T with X in R^{128x9}.  We tile it 16x16 and run
// V_WMMA_F32_16X16X4_F32 with K padded 9->12 (3 WMMAs / pixel / tile,
// 27 WMMAs per tile), keeping full f32 precision to match the reference.

typedef __attribute__((ext_vector_type(2))) float v2f;
typedef __attribute__((ext_vector_type(8))) float v8f;

#define CIN   128
#define COUT  128
#define NPAIR 8128      // 128*127/2 pairs per channel == segment size
#define NTILE 36        // upper-triangular 16x16 blocks of a 128x128 Gram

// ---------------------------------------------------------------------------
// Kernel 1: global min/max -> C1, C2; zero the 128 segment accumulators.
// ---------------------------------------------------------------------------
__global__ void ckl_prep(const float* __restrict__ x, int n,
                         float* __restrict__ stats,
                         float* __restrict__ sums,
                         float* __restrict__ counts) {
  __shared__ float smn[1024];
  __shared__ float smx[1024];
  const int tid = threadIdx.x;
  float mn = 3.402823e38f, mx = -3.402823e38f;
  for (int i = tid; i < n; i += 1024) {
    float v = x[i];
    mn = fminf(mn, v);
    mx = fmaxf(mx, v);
  }
  smn[tid] = mn;
  smx[tid] = mx;
  __syncthreads();
  for (int off = 512; off > 0; off >>= 1) {
    if (tid < off) {
      smn[tid] = fminf(smn[tid], smn[tid + off]);
      smx[tid] = fmaxf(smx[tid], smx[tid + off]);
    }
    __syncthreads();
  }
  if (tid == 0) {
    float rng = smx[0] - smn[0];
    if (rng == 0.0f) rng = 1.0f;
    float c1 = 0.01f * rng;
    float c2 = 0.03f * rng;
    stats[0] = c1 * c1;
    stats[1] = c2 * c2;
  }
  if (tid < CIN) {
    sums[tid] = 0.0f;
    counts[tid] = 0.0f;
  }
}

// ---------------------------------------------------------------------------
// Kernel 2: one wave32 per (channel, 16x16 upper-tri tile).
// ---------------------------------------------------------------------------
__global__ void __launch_bounds__(32) ckl_tiles(const float* __restrict__ x,
                                                const float* __restrict__ stats,
                                                float* __restrict__ sums,
                                                float* __restrict__ counts) {
  __shared__ float Wt[9][12];                 // 9x9 blur matrix, K-padded to 12
  __shared__ float muR[16][9], bxR[16][9];    // per-row  mu / blur(x^2)
  __shared__ float muC[16][9], bxC[16][9];    // per-col  mu / blur(x^2)

  const int tid = threadIdx.x;
  const int c = blockIdx.x / NTILE;
  int u = blockIdx.x % NTILE;
  int I = 0;
  while (u >= 8 - I) { u -= 8 - I; ++I; }     // uniform scalar loop
  const int J = I + u;                        // J >= I

  // --- 1D reflect-blur matrix W1 (win=3, sigma=1.5), then Wt = W1 (x) W1 ---
  float w1[3][3];
  {
    double g1 = exp(-1.0 / 4.5);              // gauss tap at x=+-1 (pre-norm)
    double s0 = 1.0 + 2.0 * g1;
    float a = (float)(g1 / s0);               // g[0] == g[2]
    float b = (float)(1.0 / s0);              // g[1]
    w1[0][0] = b;    w1[0][1] = 2.f * a; w1[0][2] = 0.f;
    w1[1][0] = a;    w1[1][1] = b;       w1[1][2] = a;
    w1[2][0] = 0.f;  w1[2][1] = 2.f * a; w1[2][2] = b;
  }
  for (int idx = tid; idx < 108; idx += 32) {
    int p = idx / 12, k = idx % 12;
    float v = 0.f;
    if (k < 9) v = w1[p / 3][k / 3] * w1[p % 3][k % 3];
    Wt[p][k] = v;
  }
  __syncthreads();

  // --- per-filter mu and blur(x^2) for the 16 rows (lanes 0-15) and
  //     16 cols (lanes 16-31) of this tile ---
  {
    const int half = tid >> 4, lm = tid & 15;
    const int loc = (half == 0) ? (I * 16 + lm) : (J * 16 + lm);
    const float* xf = x + (c * COUT + loc) * 9;
    float v[9];
#pragma unroll
    for (int k = 0; k < 9; ++k) v[k] = xf[k];
#pragma unroll
    for (int p = 0; p < 9; ++p) {
      float m = 0.f, q = 0.f;
#pragma unroll
      for (int k = 0; k < 9; ++k) {
        float w = Wt[p][k];
        m += w * v[k];
        q += w * v[k] * v[k];
      }
      if (half == 0) { muR[lm][p] = m; bxR[lm][p] = q; }
      else           { muC[lm][p] = m; bxC[lm][p] = q; }
    }
  }
  __syncthreads();

  // --- stage WMMA operands.  f32 16x4 A / 4x16 B layout (wave32):
  //     lanes 0-15 hold K = kc+0 (v0), kc+1 (v1); lanes 16-31 hold kc+2, kc+3.
  const int half = tid >> 4, lm = tid & 15;
  const float* xr = x + (c * COUT + I * 16 + lm) * 9;   // A row  (M = lm)
  const float* xc = x + (c * COUT + J * 16 + lm) * 9;   // B col  (N = lm)

  float ax[6], bx[6];
  int kk[6];
#pragma unroll
  for (int ch = 0; ch < 3; ++ch) {
    int k0 = ch * 4 + 2 * half;
    int k1 = k0 + 1;
    kk[2 * ch] = k0;
    kk[2 * ch + 1] = k1;
    ax[2 * ch]     = (k0 < 9) ? xr[k0] : 0.f;
    ax[2 * ch + 1] = (k1 < 9) ? xr[k1] : 0.f;
    bx[2 * ch]     = (k0 < 9) ? xc[k0] : 0.f;
    bx[2 * ch + 1] = (k1 < 9) ? xc[k1] : 0.f;
  }

  v8f acc[9] = {};                            // one 16x16 f32 tile per pixel
#pragma unroll
  for (int p = 0; p < 9; ++p) {
#pragma unroll
    for (int ch = 0; ch < 3; ++ch) {
      const int k0 = kk[2 * ch], k1 = kk[2 * ch + 1];
      v2f a, b;
      a.x = Wt[p][k0] * ax[2 * ch];           // Wt zero-padded: k<=11 safe
      a.y = Wt[p][k1] * ax[2 * ch + 1];
      b.x = bx[2 * ch];
      b.y = bx[2 * ch + 1];
      acc[p] = __builtin_amdgcn_wmma_f32_16x16x4_f32(
          false, a, false, b, (short)0, acc[p], false, false);
    }
  }

  // --- SSIM epilogue.  D layout: VGPR v, lane l -> (M = v + 8*(l>=16),
  //     N = l%16).  Segment id replicates the reference's positional
  //     repeat_interleave grouping: seg = (p_pair*128 + c) / 8128.
  const float C1 = stats[0], C2 = stats[1];
#pragma unroll
  for (int v = 0; v < 8; ++v) {
    const int i = I * 16 + v + 8 * half;      // local row filter in channel
    const int j = J * 16 + lm;                // local col filter in channel
    if (j > i) {
      float s = 0.f;
#pragma unroll
      for (int p = 0; p < 9; ++p) {
        const float m1 = muR[v + 8 * half + 0][p];
        const float m2 = muC[lm][p];
        const float b1 = bxR[v + 8 * half + 0][p];
        const float b2 = bxC[lm][p];
        const float m12 = m1 * m2;
        const float s12 = acc[p][v] - m12;
        const float s1 = b1 - m1 * m1;
        const float s2 = b2 - m2 * m2;
        const float num = (2.f * m12 + C1) * (2.f * s12 + C2);
        const float den = (m1 * m1 + m2 * m2 + C1) * (s1 + s2 + C2);
        s += num / den;
      }
      s *= (1.f / 9.f);
      const float h = s - 0.9f;               // hinge vs MARGIN
      if (h > 0.f) {
        const int pp = i * 127 - ((i * (i - 1)) >> 1) + (j - i - 1);
        const int seg = (pp * 128 + c) / NPAIR;
        atomicAdd(&sums[seg], h);
        atomicAdd(&counts[seg], 1.0f);
      }
    }
  }
}

// ---------------------------------------------------------------------------
// Kernel 3: loss = mean over 128 segments of sum / max(count, 1).
// ---------------------------------------------------------------------------
__global__ void ckl_final(const float* __restrict__ sums,
                          const float* __restrict__ counts,
                          float* __restrict__ out) {
  __shared__ float red[128];
  const int tid = threadIdx.x;
  red[tid] = sums[tid] / fmaxf(counts[tid], 1.0f);
  __syncthreads();
  for (int off = 64; off > 0; off >>= 1) {
    if (tid < off) red[tid] += red[tid + off];
    __syncthreads();
  }
  if (tid == 0) out[0] = red[0] * (1.0f / 128.0f);
}

// ---------------------------------------------------------------------------
extern "C" void kernel_launch(void* const* d_in, const int* in_sizes, int n_in,
                              void* d_out, int out_size, void* d_ws,
                              size_t ws_size, hipStream_t stream) {
  (void)n_in; (void)out_size; (void)ws_size;
  const float* x = (const float*)d_in[0];     // (128,128,3,3) f32
  float* ws = (float*)d_ws;
  float* stats = ws;                          // [0]=C1, [1]=C2
  float* sums = ws + 8;                       // 128 segment hinge sums
  float* counts = ws + 8 + CIN;               // 128 segment positive counts
  const int n = in_sizes[0];

  ckl_prep<<<1, 1024, 0, stream>>>(x, n, stats, sums, counts);
  ckl_tiles<<<CIN * NTILE, 32, 0, stream>>>(x, stats, sums, counts);
  ckl_final<<<1, CIN, 0, stream>>>(sums, counts, (float*)d_out);
}